// SchemaGCN_61151744361083
// MI455X (gfx1250) — compile-verified
//
#include <hip/hip_runtime.h>
#include <hip/hip_bf16.h>

typedef float v2f __attribute__((ext_vector_type(2)));
typedef float v8f __attribute__((ext_vector_type(8)));

#define HID 64

// ---------------------------------------------------------------------------
// Degree / normalization kernels
// ---------------------------------------------------------------------------
__global__ void deg_init_kernel(float* __restrict__ deg, int n) {
    int i = blockIdx.x * blockDim.x + threadIdx.x;
    if (i < n) deg[i] = 1.0f;               // self-loop contributes 1
}

__global__ void deg_count_kernel(const int* __restrict__ dst,
                                 float* __restrict__ deg, int e) {
    int i = blockIdx.x * blockDim.x + threadIdx.x;
    if (i < e) atomicAdd(&deg[dst[i]], 1.0f);
}

__global__ void deg_finalize_kernel(float* __restrict__ dinv, int n) {
    int i = blockIdx.x * blockDim.x + threadIdx.x;
    if (i < n) dinv[i] = rsqrtf(dinv[i]);   // deg >= 1 always
}

// ---------------------------------------------------------------------------
// WMMA fp32 GEMM:  Y[nrows, 64] = X[nrows, K] @ W[K, 64]  (+ bias, + relu)
// One wave computes a 16-row x 64-col tile using V_WMMA_F32_16X16X4_F32.
// A layout (16x4 f32): lanes 0-15 -> M=lane, VGPR0/1 = K=k0,k0+1;
//                      lanes 16-31 -> M=lane-16, VGPR0/1 = K=k0+2,k0+3.
// B layout (4x16 f32): VGPR0 = rows K=k0 (lanes 0-15) / k0+2 (lanes 16-31),
//                      VGPR1 = rows K=k0+1 / k0+3; col = lane%16.
// C/D layout: VGPR j -> row = j (lanes 0-15) / j+8 (lanes 16-31), col = lane%16.
// ---------------------------------------------------------------------------
template<int K, bool BIAS, bool RELU>
__global__ void wmma_gemm_kernel(const float* __restrict__ X,
                                 const float* __restrict__ W,
                                 const float* __restrict__ bias,
                                 float* __restrict__ Y,
                                 int nrows)
{
    __shared__ float sW[K * HID];
    for (int i = threadIdx.x; i < K * HID; i += blockDim.x) sW[i] = W[i];
    __syncthreads();

    const int lane  = threadIdx.x & 31;
    const int wave  = threadIdx.x >> 5;
    const int waves = blockDim.x >> 5;
    const int row0  = (blockIdx.x * waves + wave) * 16;
    if (row0 >= nrows) return;              // wave-uniform: EXEC stays all-1s

    const int m     = lane & 15;
    const int khalf = (lane >> 4) << 1;     // 0 for lanes 0-15, 2 for 16-31

    v8f acc[4] = {};
    const float* xrow = X + (size_t)(row0 + m) * K;

    for (int k0 = 0; k0 < K; k0 += 4) {
        v2f a = *(const v2f*)(xrow + k0 + khalf);   // 8B aligned
#pragma unroll
        for (int c = 0; c < 4; ++c) {
            v2f b;
            b.x = sW[(k0 + khalf    ) * HID + c * 16 + m];
            b.y = sW[(k0 + khalf + 1) * HID + c * 16 + m];
            acc[c] = __builtin_amdgcn_wmma_f32_16x16x4_f32(
                false, a, false, b, (short)0, acc[c], false, false);
        }
    }

    const int rbase = row0 + ((lane >> 4) << 3);    // +8 for upper-half lanes
#pragma unroll
    for (int c = 0; c < 4; ++c) {
        const int col = c * 16 + m;
        const float bv = BIAS ? bias[col] : 0.0f;
#pragma unroll
        for (int j = 0; j < 8; ++j) {
            float v = acc[c][j] + bv;
            if (RELU) v = fmaxf(v, 0.0f);
            Y[(size_t)(rbase + j) * HID + col] = v;
        }
    }
}

// ---------------------------------------------------------------------------
// agg[i][f] = h[i][f] * dinv[i]^2   (self-loop message, also zero-bases agg)
// ---------------------------------------------------------------------------
__global__ void init_agg_kernel(const float* __restrict__ h,
                                const float* __restrict__ dinv,
                                float* __restrict__ agg, int total) {
    int i = blockIdx.x * blockDim.x + threadIdx.x;
    if (i < total) {
        float dv = dinv[i >> 6];
        agg[i] = h[i] * dv * dv;
    }
}

// ---------------------------------------------------------------------------
// One wave per edge: gather h[src] row (256B coalesced), scale by
// dinv[src]*dinv[dst], scatter-add into agg[dst] with global_atomic_add_f32.
// ---------------------------------------------------------------------------
__global__ void scatter_edges_kernel(const int* __restrict__ src,
                                     const int* __restrict__ dst,
                                     const float* __restrict__ dinv,
                                     const float* __restrict__ h,
                                     float* __restrict__ agg, int e)
{
    int edge = blockIdx.x * (blockDim.x >> 5) + (threadIdx.x >> 5);
    if (edge >= e) return;
    int lane = threadIdx.x & 31;
    int s = src[edge];
    int d = dst[edge];
    float norm = dinv[s] * dinv[d];
    const float* hs = h + (size_t)s * HID;
    float v0 = hs[lane]      * norm;
    float v1 = hs[lane + 32] * norm;
    float* ad = agg + (size_t)d * HID;
    atomicAdd(&ad[lane],      v0);
    atomicAdd(&ad[lane + 32], v1);
}

// ---------------------------------------------------------------------------
// y[i] = relu(agg[i] + b[i % 64])
// ---------------------------------------------------------------------------
__global__ void finalize_kernel(const float* __restrict__ agg,
                                const float* __restrict__ b,
                                float* __restrict__ y, int total) {
    int i = blockIdx.x * blockDim.x + threadIdx.x;
    if (i < total) y[i] = fmaxf(agg[i] + b[i & 63], 0.0f);
}

// ---------------------------------------------------------------------------
extern "C" void kernel_launch(void* const* d_in, const int* in_sizes, int n_in,
                              void* d_out, int out_size, void* d_ws, size_t ws_size,
                              hipStream_t stream)
{
    const float* x     = (const float*)d_in[0];   // [N,128]
    const int*   eidx  = (const int*)  d_in[1];   // [2,E]
    const float* W_pre = (const float*)d_in[2];   // [128,64]
    const float* b_pre = (const float*)d_in[3];   // [64]
    const float* W1    = (const float*)d_in[4];   // [64,64]
    const float* b1    = (const float*)d_in[5];
    const float* W2    = (const float*)d_in[6];
    const float* b2    = (const float*)d_in[7];

    const int IN = 128;
    const int N  = in_sizes[0] / IN;
    const int E  = in_sizes[1] / 2;
    const int* src = eidx;
    const int* dst = eidx + E;

    float* out_x   = (float*)d_out;                 // first half: final x
    float* out_ori = (float*)d_out + (size_t)N * HID; // second half: ori_x

    float* dinv = (float*)d_ws;                     // N
    float* h    = dinv + N;                         // N*64
    float* agg  = h + (size_t)N * HID;              // N*64

    const int NH = N * HID;
    const int TPB = 256;
    const int WPB = TPB / 32;
    const int gemm_blocks = ((N / 16) + WPB - 1) / WPB;
    const int edge_blocks = (E + WPB - 1) / WPB;

    // ---- degree normalization -------------------------------------------
    deg_init_kernel<<<(N + TPB - 1) / TPB, TPB, 0, stream>>>(dinv, N);
    deg_count_kernel<<<(E + TPB - 1) / TPB, TPB, 0, stream>>>(dst, dinv, E);
    deg_finalize_kernel<<<(N + TPB - 1) / TPB, TPB, 0, stream>>>(dinv, N);

    // ---- ori_x = x @ W_pre + b_pre  (written straight into d_out) -------
    wmma_gemm_kernel<128, true, false><<<gemm_blocks, TPB, 0, stream>>>(
        x, W_pre, b_pre, out_ori, N);

    // ---- layer 1 ---------------------------------------------------------
    wmma_gemm_kernel<64, false, false><<<gemm_blocks, TPB, 0, stream>>>(
        out_ori, W1, nullptr, h, N);
    init_agg_kernel<<<(NH + TPB - 1) / TPB, TPB, 0, stream>>>(h, dinv, agg, NH);
    scatter_edges_kernel<<<edge_blocks, TPB, 0, stream>>>(src, dst, dinv, h, agg, E);
    finalize_kernel<<<(NH + TPB - 1) / TPB, TPB, 0, stream>>>(agg, b1, out_x, NH);

    // ---- layer 2 (reads layer-1 activations from d_out first half, then
    //      overwrites that half with the final result) ---------------------
    wmma_gemm_kernel<64, false, false><<<gemm_blocks, TPB, 0, stream>>>(
        out_x, W2, nullptr, h, N);
    init_agg_kernel<<<(NH + TPB - 1) / TPB, TPB, 0, stream>>>(h, dinv, agg, NH);
    scatter_edges_kernel<<<edge_blocks, TPB, 0, stream>>>(src, dst, dinv, h, agg, E);
    finalize_kernel<<<(NH + TPB - 1) / TPB, TPB, 0, stream>>>(agg, b2, out_x, NH);
}